// GATLayer_66176856097229
// MI455X (gfx1250) — compile-verified
//
#include <hip/hip_runtime.h>

typedef __attribute__((ext_vector_type(2))) float v2f;
typedef __attribute__((ext_vector_type(8))) float v8f;

#define HF    128   // IN_F == H*F == 128
#define NHEAD 8
#define FDIM  16

// ---- order-preserving float<->uint encoding so atomicMax(u32) == float max ----
__device__ __forceinline__ unsigned fenc(float f) {
  unsigned u = __float_as_uint(f);
  return (u & 0x80000000u) ? ~u : (u | 0x80000000u);
}
__device__ __forceinline__ float fdec(unsigned u) {
  return __uint_as_float((u & 0x80000000u) ? (u ^ 0x80000000u) : ~u);
}

// =====================================================================
// Kernel 1: h = x @ Wv + bv   (N x 128) = (N x 128)(128 x 128)
// One block per 16-row tile; wave w computes the 16-col tile n0 = 16*w.
// V_WMMA_F32_16X16X4_F32, K looped 128/4 = 32 times (unroll 8).
// A layout: lanes 0-15 -> M=lane, VGPR{0,1}=K{0,1}; lanes 16-31 -> K{2,3}.
// B layout: lanes 0-15 -> N=lane, VGPR{0,1}=K{0,1}; lanes 16-31 -> K{2,3}.
// C layout: VGPR r -> row M=r (lanes 0-15) / M=r+8 (lanes 16-31), N=lane&15.
// =====================================================================
__global__ __launch_bounds__(256) void gemm_hflat(
    const float* __restrict__ x, const float* __restrict__ Wv,
    const float* __restrict__ bv, float* __restrict__ h, int n) {
  const int lane = threadIdx.x & 31;
  const int wave = threadIdx.x >> 5;
  const int m0 = blockIdx.x << 4;
  const int n0 = wave << 4;
  const int cr = lane & 15;          // A-row / B-col / C-col
  const int kh = (lane >> 4) << 1;   // K sub-offset: 0 or 2
  int arow = m0 + cr; if (arow >= n) arow = n - 1;
  const float* __restrict__ xr = x + (size_t)arow * HF;
  v8f c = {};
  #pragma unroll 8
  for (int kk = 0; kk < HF; kk += 4) {
    v2f a, b;
    a.x = xr[kk + kh];
    a.y = xr[kk + kh + 1];
    b.x = Wv[(size_t)(kk + kh) * HF + n0 + cr];
    b.y = Wv[(size_t)(kk + kh + 1) * HF + n0 + cr];
    c = __builtin_amdgcn_wmma_f32_16x16x4_f32(false, a, false, b,
                                              (short)0, c, false, false);
  }
  const float bias = bv[n0 + cr];
  const int rbase = m0 + ((lane >> 4) << 3);
  float* __restrict__ hp = h + (size_t)rbase * HF + n0 + cr;
  if (m0 + 16 <= n) {                // wave-uniform full-tile fast path
    #pragma unroll
    for (int r = 0; r < 8; ++r) hp[(size_t)r * HF] = c[r] + bias;
  } else {
    for (int r = 0; r < 8; ++r)
      if (rbase + r < n) hp[(size_t)r * HF] = c[r] + bias;
  }
}

// =====================================================================
// Kernel 2: [q | k] = h @ [Wq | Wk] + [bq | bk]
// Wq,Wk are 128x8 -> concatenated they form one 128x16 B tile.
// One wave per 16-node tile (8 waves per block).
// =====================================================================
__global__ __launch_bounds__(256) void qk_wmma(
    const float* __restrict__ h, const float* __restrict__ Wq,
    const float* __restrict__ bq, const float* __restrict__ Wk,
    const float* __restrict__ bk, float* __restrict__ q,
    float* __restrict__ k, int n, int mtiles) {
  const int mt = blockIdx.x * 8 + (threadIdx.x >> 5);
  if (mt >= mtiles) return;          // wave-uniform: EXEC stays all-ones
  const int lane = threadIdx.x & 31;
  const int m0 = mt << 4;
  const int cr = lane & 15;
  const int kh = (lane >> 4) << 1;
  int arow = m0 + cr; if (arow >= n) arow = n - 1;
  const float* __restrict__ hr = h + (size_t)arow * HF;
  const float* __restrict__ W0 = (cr < 8) ? Wq : Wk;
  const int c8 = cr & 7;
  v8f c = {};
  #pragma unroll 8
  for (int kk = 0; kk < HF; kk += 4) {
    v2f a, b;
    a.x = hr[kk + kh];
    a.y = hr[kk + kh + 1];
    b.x = W0[(size_t)(kk + kh) * NHEAD + c8];
    b.y = W0[(size_t)(kk + kh + 1) * NHEAD + c8];
    c = __builtin_amdgcn_wmma_f32_16x16x4_f32(false, a, false, b,
                                              (short)0, c, false, false);
  }
  const float bias = (cr < 8) ? bq[c8] : bk[c8];
  float* __restrict__ outp = ((cr < 8) ? q : k) + c8;
  const int rbase = m0 + ((lane >> 4) << 3);
  if (m0 + 16 <= n) {                // wave-uniform full-tile fast path
    #pragma unroll
    for (int r = 0; r < 8; ++r)
      outp[(size_t)(rbase + r) * NHEAD] = c[r] + bias;
  } else {
    for (int r = 0; r < 8; ++r)
      if (rbase + r < n) outp[(size_t)(rbase + r) * NHEAD] = c[r] + bias;
  }
}

// =====================================================================
// Kernel 3: zero out[ N*16 ], denom[ N*8 ], segmax[ N*8 ] (0 < enc(-inf))
// =====================================================================
__global__ __launch_bounds__(256) void init_buffers(
    float* __restrict__ out, float* __restrict__ denom,
    unsigned* __restrict__ smax, int n) {
  int t = blockIdx.x * 256 + threadIdx.x;
  if (t < n * FDIM) out[t] = 0.0f;
  if (t < n * NHEAD) { denom[t] = 0.0f; smax[t] = 0u; }
}

// =====================================================================
// Kernel 4: per (edge, head): leaky-relu logit; atomic max into segmax[dst,h]
// =====================================================================
__global__ __launch_bounds__(256) void edge_logit_max(
    const int* __restrict__ src, const int* __restrict__ dst,
    const float* __restrict__ qb, const float* __restrict__ kb,
    float* __restrict__ ecoef, unsigned* __restrict__ smax, int tot) {
  int t = blockIdx.x * 256 + threadIdx.x;
  if (t >= tot) return;
  const int e = t >> 3, hh = t & 7;
  const int s = src[e], d = dst[e];
  float v = qb[(size_t)s * NHEAD + hh] + kb[(size_t)d * NHEAD + hh];
  v = (v >= 0.0f) ? v : 0.2f * v;
  ecoef[t] = v;
  atomicMax(&smax[(size_t)d * NHEAD + hh], fenc(v));
}

// =====================================================================
// Kernel 5: per (edge, head): ex = exp(logit - max); atomic add into denom
// =====================================================================
__global__ __launch_bounds__(256) void edge_exp_sum(
    const int* __restrict__ dst, float* __restrict__ ecoef,
    const unsigned* __restrict__ smax, float* __restrict__ denom, int tot) {
  int t = blockIdx.x * 256 + threadIdx.x;
  if (t >= tot) return;
  const int e = t >> 3, hh = t & 7;
  const int d = dst[e];
  float m = fdec(smax[(size_t)d * NHEAD + hh]);
  float ex = __expf(ecoef[t] - m);
  ecoef[t] = ex;
  atomicAdd(&denom[(size_t)d * NHEAD + hh], ex);
}

// =====================================================================
// Kernel 5.5: denom -> 1/denom in place (400k elems), so the 12.8M-thread
// scatter kernel multiplies instead of dividing 8x per thread.
// Safe across graph replays: init_buffers re-zeroes denom every launch.
// =====================================================================
__global__ __launch_bounds__(256) void recip_denom(
    float* __restrict__ denom, int tot) {
  int t = blockIdx.x * 256 + threadIdx.x;
  if (t >= tot) return;
  float d = denom[t];
  denom[t] = (d != 0.0f) ? (1.0f / d) : 0.0f;   // empty segments never read
}

// =====================================================================
// Kernel 6: per (edge, f): acc = sum_h attn[e,h]*h[src,h,f];
//           atomicAdd out[dst,f] += acc/H.  16 lanes of an edge read
//           contiguous 64B per head from h[src] (coalesced gather).
// =====================================================================
__global__ __launch_bounds__(256) void edge_scatter(
    const int* __restrict__ src, const int* __restrict__ dst,
    const float* __restrict__ h, const float* __restrict__ ecoef,
    const float* __restrict__ rdenom, float* __restrict__ out, int tot) {
  int t = blockIdx.x * 256 + threadIdx.x;
  if (t >= tot) return;
  const int e = t >> 4, f = t & 15;
  const int s = src[e], d = dst[e];
  const float* __restrict__ hs = h + (size_t)s * HF + f;
  const float* __restrict__ ec = ecoef + (size_t)e * NHEAD;
  const float* __restrict__ rd = rdenom + (size_t)d * NHEAD;
  float acc = 0.0f;
  #pragma unroll
  for (int hh = 0; hh < NHEAD; ++hh) {
    float attn = ec[hh] * rd[hh];
    acc += attn * hs[hh * FDIM];
  }
  atomicAdd(&out[(size_t)d * FDIM + f], acc * (1.0f / (float)NHEAD));
}

// =====================================================================
extern "C" void kernel_launch(void* const* d_in, const int* in_sizes, int n_in,
                              void* d_out, int out_size, void* d_ws, size_t ws_size,
                              hipStream_t stream) {
  const float* x  = (const float*)d_in[0];
  const int*   src = (const int*)d_in[1];
  const int*   dst = (const int*)d_in[2];
  const float* Wv = (const float*)d_in[3];
  const float* bv = (const float*)d_in[4];
  const float* Wq = (const float*)d_in[5];
  const float* bq = (const float*)d_in[6];
  const float* Wk = (const float*)d_in[7];
  const float* bk = (const float*)d_in[8];
  float* out = (float*)d_out;

  const int n = in_sizes[0] / HF;   // 50000
  const int e = in_sizes[1];        // 800000

  // workspace carve-up (fp32 unless noted)
  float*    h     = (float*)d_ws;                     // n*128
  float*    qb    = h + (size_t)n * HF;               // n*8
  float*    kb    = qb + (size_t)n * NHEAD;           // n*8
  float*    denom = kb + (size_t)n * NHEAD;           // n*8
  unsigned* smax  = (unsigned*)(denom + (size_t)n * NHEAD); // n*8 (u32)
  float*    ecoef = (float*)(smax + (size_t)n * NHEAD);     // e*8

  const int mtiles = (n + 15) / 16;

  gemm_hflat<<<dim3(mtiles), dim3(256), 0, stream>>>(x, Wv, bv, h, n);
  qk_wmma<<<dim3((mtiles + 7) / 8), dim3(256), 0, stream>>>(h, Wq, bq, Wk, bk,
                                                            qb, kb, n, mtiles);
  const int tInit = n * FDIM;
  init_buffers<<<dim3((tInit + 255) / 256), dim3(256), 0, stream>>>(out, denom,
                                                                    smax, n);
  const int tEH = e * NHEAD;
  edge_logit_max<<<dim3((tEH + 255) / 256), dim3(256), 0, stream>>>(
      src, dst, qb, kb, ecoef, smax, tEH);
  edge_exp_sum<<<dim3((tEH + 255) / 256), dim3(256), 0, stream>>>(
      dst, ecoef, smax, denom, tEH);
  const int tND = n * NHEAD;
  recip_denom<<<dim3((tND + 255) / 256), dim3(256), 0, stream>>>(denom, tND);
  const int tSC = e * FDIM;
  edge_scatter<<<dim3((tSC + 255) / 256), dim3(256), 0, stream>>>(
      src, dst, h, ecoef, denom, out, tSC);
}